// GCN1_23192823399146
// MI455X (gfx1250) — compile-verified
//
#include <hip/hip_runtime.h>

#define IN_F  128
#define HID_F 128
#define OUT_F 64

typedef __attribute__((ext_vector_type(2))) float v2f;
typedef __attribute__((ext_vector_type(8))) float v8f;

// ---------------- elementwise utility kernels ----------------

__global__ void fill_kernel(float* __restrict__ p, float v, int n) {
  int i = blockIdx.x * blockDim.x + threadIdx.x;
  if (i < n) p[i] = v;
}

__global__ void deg_kernel(const int* __restrict__ col, float* __restrict__ deg, int E) {
  int e = blockIdx.x * blockDim.x + threadIdx.x;
  if (e < E) atomicAdd(&deg[col[e]], 1.0f);
}

__global__ void dinv_kernel(float* __restrict__ deg, int n) {
  int i = blockIdx.x * blockDim.x + threadIdx.x;
  if (i < n) deg[i] = rsqrtf(deg[i]);   // deg >= 1 (self loop), so no zero guard needed
}

// h = relu(agg + xw * dinv^2 + b1), written in place over agg
__global__ void bias_relu_kernel(const float* __restrict__ xw,
                                 float* __restrict__ agg,
                                 const float* __restrict__ dinv,
                                 const float* __restrict__ b1,
                                 int total) {
  int i = blockIdx.x * blockDim.x + threadIdx.x;
  if (i >= total) return;
  int node = i >> 7;          // /128
  int f    = i & 127;
  float di = dinv[node];
  float v  = agg[i] + xw[i] * di * di + b1[f];
  agg[i] = fmaxf(v, 0.0f);
}

// out += hw * dinv^2 + b2   (out already holds edge-aggregated sum)
__global__ void final_kernel(const float* __restrict__ hw,
                             const float* __restrict__ dinv,
                             const float* __restrict__ b2,
                             float* __restrict__ out,
                             int total) {
  int i = blockIdx.x * blockDim.x + threadIdx.x;
  if (i >= total) return;
  int node = i >> 6;          // /64
  int f    = i & 63;
  float di = dinv[node];
  out[i] = out[i] + hw[i] * di * di + b2[f];
}

// ---------------- WMMA GEMM: C[nrows x F] = A[nrows x 128] @ B[128 x F] ----------------
// One 16x16 output tile per wave, exact fp32 via V_WMMA_F32_16X16X4_F32.
// Block = (F/16) waves covering 16 output columns each; grid.x = row tiles.

template <int F>
__global__ void __launch_bounds__((F / 16) * 32)
gemm_wmma_kernel(const float* __restrict__ A,
                 const float* __restrict__ B,
                 float* __restrict__ C,
                 int nrows) {
  const int lane = threadIdx.x & 31;
  const int wave = threadIdx.x >> 5;
  const int row0 = blockIdx.x * 16;
  if (row0 >= nrows) return;                 // uniform per block; EXEC stays all-1 for WMMA
  const int col0 = wave * 16;
  const int m    = lane & 15;                // M (for A) / N (for B,D) index
  const int half = lane >> 4;                // K sub-phase selector

  v8f acc = {};
  const float* arow = A + (size_t)(row0 + m) * 128;

#pragma unroll
  for (int k = 0; k < 128; k += 4) {
    const int ka = k + 2 * half;
    v2f a, b;
    a.x = arow[ka];                          // A[m][ka],  A 16x4 fragment: 2 VGPRs/lane
    a.y = arow[ka + 1];                      // A[m][ka+1]
    b.x = B[(size_t)ka * F + col0 + m];      // B[ka][col0+m], row-striped across lanes
    b.y = B[(size_t)(ka + 1) * F + col0 + m];
    acc = __builtin_amdgcn_wmma_f32_16x16x4_f32(
        /*neg_a=*/false, a, /*neg_b=*/false, b,
        /*c_mod=*/(short)0, acc, /*reuse_a=*/false, /*reuse_b=*/false);
  }

  // D layout: VGPR v -> row (v + 8*half), col (lane&15)
#pragma unroll
  for (int v = 0; v < 8; ++v) {
    C[(size_t)(row0 + v + 8 * half) * F + col0 + m] = acc[v];
  }
}

// ---------------- edge scatter (the memory-bound hot loop) ----------------
// One wave per edge; float4 gather, scaled, 4 f32 atomics per lane. 128 feats.
__global__ void scatter128_kernel(const int* __restrict__ row,
                                  const int* __restrict__ col,
                                  const float* __restrict__ dinv,
                                  const float* __restrict__ xw,
                                  float* __restrict__ agg,
                                  int E) {
  int gw   = (blockIdx.x * blockDim.x + threadIdx.x) >> 5;
  int lane = threadIdx.x & 31;
  if (gw >= E) return;
  int r = row[gw];
  int c = col[gw];
  float nrm = dinv[r] * dinv[c];
  const float4* src = (const float4*)(xw + (size_t)r * 128);
  float4 v = src[lane];
  float* dst = agg + (size_t)c * 128 + lane * 4;
  atomicAdd(dst + 0, v.x * nrm);
  atomicAdd(dst + 1, v.y * nrm);
  atomicAdd(dst + 2, v.z * nrm);
  atomicAdd(dst + 3, v.w * nrm);
}

// 64-feature version: float2 per lane.
__global__ void scatter64_kernel(const int* __restrict__ row,
                                 const int* __restrict__ col,
                                 const float* __restrict__ dinv,
                                 const float* __restrict__ hw,
                                 float* __restrict__ agg,
                                 int E) {
  int gw   = (blockIdx.x * blockDim.x + threadIdx.x) >> 5;
  int lane = threadIdx.x & 31;
  if (gw >= E) return;
  int r = row[gw];
  int c = col[gw];
  float nrm = dinv[r] * dinv[c];
  const float2* src = (const float2*)(hw + (size_t)r * 64);
  float2 v = src[lane];
  float* dst = agg + (size_t)c * 64 + lane * 2;
  atomicAdd(dst + 0, v.x * nrm);
  atomicAdd(dst + 1, v.y * nrm);
}

// ---------------- launcher ----------------

extern "C" void kernel_launch(void* const* d_in, const int* in_sizes, int n_in,
                              void* d_out, int out_size, void* d_ws, size_t ws_size,
                              hipStream_t stream) {
  const float* x  = (const float*)d_in[0];
  const int*   ei = (const int*)d_in[1];     // [2, E]
  const float* W1 = (const float*)d_in[2];   // [128,128]
  const float* b1 = (const float*)d_in[3];   // [128]
  const float* W2 = (const float*)d_in[4];   // [128,64]
  const float* b2 = (const float*)d_in[5];   // [64]
  float* out = (float*)d_out;                // [N,64]

  const int N = in_sizes[0] / IN_F;
  const int E = in_sizes[1] / 2;
  const int* row = ei;                       // sources
  const int* col = ei + E;                   // targets

  // workspace layout (floats): xw[N*128] | agg1/h[N*128] | hw[N*64] | dinv[N]
  float* ws   = (float*)d_ws;
  float* xw   = ws;
  float* agg1 = xw + (size_t)N * 128;
  float* hw   = agg1 + (size_t)N * 128;
  float* deg  = hw + (size_t)N * 64;         // becomes dinv in place

  const int n128 = N * 128;
  const int n64  = N * 64;

  // init accumulators + degrees (deg starts at 1.0 for the self loop)
  fill_kernel<<<(n128 + 255) / 256, 256, 0, stream>>>(agg1, 0.0f, n128);
  fill_kernel<<<(n64  + 255) / 256, 256, 0, stream>>>(out,  0.0f, n64);
  fill_kernel<<<(N    + 255) / 256, 256, 0, stream>>>(deg,  1.0f, N);

  // degree -> dinv
  deg_kernel<<<(E + 255) / 256, 256, 0, stream>>>(col, deg, E);
  dinv_kernel<<<(N + 255) / 256, 256, 0, stream>>>(deg, N);

  // layer 1: xw = X @ W1 (WMMA), edge scatter, self-loop + bias + relu
  gemm_wmma_kernel<128><<<(N + 15) / 16, 256, 0, stream>>>(x, W1, xw, N);
  scatter128_kernel<<<(E + 7) / 8, 256, 0, stream>>>(row, col, deg, xw, agg1, E);
  bias_relu_kernel<<<(n128 + 255) / 256, 256, 0, stream>>>(xw, agg1, deg, b1, n128);

  // layer 2: hw = h @ W2 (WMMA), edge scatter into out, self-loop + bias
  gemm_wmma_kernel<64><<<(N + 15) / 16, 128, 0, stream>>>(agg1, W2, hw, N);
  scatter64_kernel<<<(E + 7) / 8, 256, 0, stream>>>(row, col, deg, hw, out, E);
  final_kernel<<<(n64 + 255) / 256, 256, 0, stream>>>(hw, deg, b2, out, n64);
}